// GCN_9887014715508
// MI455X (gfx1250) — compile-verified
//
#include <hip/hip_runtime.h>
#include <math.h>

#define N_NODES 169343
#define N_EDGESC 2332486
#define FDIM 128
#define ODIM 40
#define BN_EPS 1e-5f

typedef float v2f __attribute__((ext_vector_type(2)));
typedef float v8f __attribute__((ext_vector_type(8)));

// ------------------------------------------------------------------
// degree / normalization
// ------------------------------------------------------------------
__global__ void init_deg_kernel(float* deg, int n) {
  int i = blockIdx.x * blockDim.x + threadIdx.x;
  if (i < n) deg[i] = 1.0f;  // self-loop contributes 1 to every node's degree
}

__global__ void degree_kernel(const int* __restrict__ col, float* __restrict__ deg, int E) {
  int e = blockIdx.x * blockDim.x + threadIdx.x;
  if (e < E) atomicAdd(&deg[col[e]], 1.0f);
}

__global__ void dinv_kernel(float* deg, int n) {
  int i = blockIdx.x * blockDim.x + threadIdx.x;
  if (i < n) deg[i] = rsqrtf(deg[i]);  // deg >= 1 always (self-loop)
}

// ------------------------------------------------------------------
// fp32 WMMA GEMM:  C[M x NC] = A[M x K] * B[K x NC]
// One wave computes one 16x16 output tile via V_WMMA_F32_16X16X4_F32.
// NT = number of 16-wide column tiles = waves per block; grid.x = M tiles.
// ------------------------------------------------------------------
template <int NC, int NT>
__global__ __launch_bounds__(32 * NT)
void gemm_wmma_f32(const float* __restrict__ A, const float* __restrict__ B,
                   float* __restrict__ C, int M, int K) {
  const int wave = threadIdx.x >> 5;
  const int lane = threadIdx.x & 31;
  const int half = lane >> 4;     // 0: lanes 0-15, 1: lanes 16-31
  const int l15  = lane & 15;

  const int m0 = blockIdx.x * 16;
  const int n0 = wave * 16;

  // clamp OOB rows/cols for loads so EXEC stays all-ones for WMMA;
  // garbage only feeds rows/cols that are masked at store time.
  int arow = m0 + l15;
  if (arow >= M) arow = M - 1;
  int bcol = n0 + l15;
  if (bcol >= NC) bcol = NC - 1;

  v8f acc = {};
#pragma unroll 4
  for (int k = 0; k < K; k += 4) {
    // A 16x4 tile (fp32): VGPR0/1 hold K = {k,k+1} for lanes 0-15, {k+2,k+3} for lanes 16-31
    const int ka = k + (half << 1);
    const float2 av = *reinterpret_cast<const float2*>(A + (size_t)arow * K + ka);
    v2f a = {av.x, av.y};
    // B 4x16 tile: rows striped across lanes (mirror of A layout)
    v2f b = {B[(size_t)ka * NC + bcol], B[(size_t)(ka + 1) * NC + bcol]};
    acc = __builtin_amdgcn_wmma_f32_16x16x4_f32(
        /*neg_a=*/false, a, /*neg_b=*/false, b,
        /*c_mod=*/(short)0, acc, /*reuse_a=*/false, /*reuse_b=*/false);
  }

  // C/D layout: VGPR r -> row m0 + r + 8*half, col n0 + l15
#pragma unroll
  for (int r = 0; r < 8; ++r) {
    const int row = m0 + r + (half << 3);
    const int colI = n0 + l15;
    if (row < M && colI < NC) C[(size_t)row * NC + colI] = acc[r];
  }
}

// ------------------------------------------------------------------
// Edge scatter (message passing): agg[col] += dinv[row]*dinv[col] * t[row]
// One wave per edge; 128 feats = 32 lanes x float4, 40 feats = 20 lanes x float2.
// ------------------------------------------------------------------
__global__ __launch_bounds__(256)
void scatter128_kernel(const int* __restrict__ rowI, const int* __restrict__ colI,
                       const float* __restrict__ dinv, const float* __restrict__ t,
                       float* __restrict__ agg, int E) {
  const int e = (int)((blockIdx.x * (size_t)blockDim.x + threadIdx.x) >> 5);
  const int lane = threadIdx.x & 31;
  if (e >= E) return;
  const int r = rowI[e], c = colI[e];
  const float nrm = dinv[r] * dinv[c];
  const float4 v = reinterpret_cast<const float4*>(t + (size_t)r * FDIM)[lane];
  float* dst = agg + (size_t)c * FDIM + lane * 4;
  atomicAdd(dst + 0, nrm * v.x);
  atomicAdd(dst + 1, nrm * v.y);
  atomicAdd(dst + 2, nrm * v.z);
  atomicAdd(dst + 3, nrm * v.w);
}

__global__ __launch_bounds__(256)
void scatter40_kernel(const int* __restrict__ rowI, const int* __restrict__ colI,
                      const float* __restrict__ dinv, const float* __restrict__ t,
                      float* __restrict__ agg, int E) {
  const int e = (int)((blockIdx.x * (size_t)blockDim.x + threadIdx.x) >> 5);
  const int lane = threadIdx.x & 31;
  if (e >= E) return;
  const int r = rowI[e], c = colI[e];
  const float nrm = dinv[r] * dinv[c];
  if (lane < 20) {
    const float2 v = reinterpret_cast<const float2*>(t + (size_t)r * ODIM)[lane];
    float* dst = agg + (size_t)c * ODIM + lane * 2;
    atomicAdd(dst + 0, nrm * v.x);
    atomicAdd(dst + 1, nrm * v.y);
  }
}

// self-loop contribution + conv bias (runs after edge scatter; no atomics needed)
template <int F>
__global__ void selfloop_bias_kernel(const float* __restrict__ t, const float* __restrict__ dinv,
                                     const float* __restrict__ bias, float* __restrict__ agg,
                                     int n) {
  const size_t idx = (size_t)blockIdx.x * blockDim.x + threadIdx.x;
  if (idx >= (size_t)n * F) return;
  const int i = (int)(idx / F);
  const int f = (int)(idx % F);
  const float di = dinv[i];
  agg[idx] += di * di * t[idx] + bias[f];
}

// ------------------------------------------------------------------
// BatchNorm (batch stats over node dim, biased variance) + ReLU
// ------------------------------------------------------------------
__global__ __launch_bounds__(FDIM)
void bn_stats_kernel(const float* __restrict__ h, double* __restrict__ sum,
                     double* __restrict__ sumsq, int n, int rowsPerBlock) {
  const int f = threadIdx.x;
  const int r0 = blockIdx.x * rowsPerBlock;
  int r1 = r0 + rowsPerBlock;
  if (r1 > n) r1 = n;
  float s = 0.f, q = 0.f;
  for (int r = r0; r < r1; ++r) {
    const float v = h[(size_t)r * FDIM + f];
    s += v;
    q += v * v;
  }
  atomicAdd(&sum[f], (double)s);
  atomicAdd(&sumsq[f], (double)q);
}

__global__ __launch_bounds__(FDIM)
void bn_finalize_kernel(const double* __restrict__ sum, const double* __restrict__ sumsq,
                        const float* __restrict__ gamma, const float* __restrict__ beta,
                        float* __restrict__ scale, float* __restrict__ shift, int n) {
  const int f = threadIdx.x;
  const double mu = sum[f] / n;
  const double var = sumsq[f] / n - mu * mu;
  const float sc = gamma[f] * rsqrtf((float)var + BN_EPS);
  scale[f] = sc;
  shift[f] = beta[f] - (float)mu * sc;
}

__global__ void bn_relu_kernel(float* __restrict__ h, const float* __restrict__ scale,
                               const float* __restrict__ shift, size_t total) {
  const size_t idx = (size_t)blockIdx.x * blockDim.x + threadIdx.x;
  if (idx >= total) return;
  const int f = (int)(idx & (FDIM - 1));
  const float v = h[idx] * scale[f] + shift[f];
  h[idx] = v > 0.f ? v : 0.f;
}

// ------------------------------------------------------------------
// log_softmax over 40 classes, one wave per row (wave32 shuffles)
// ------------------------------------------------------------------
__global__ __launch_bounds__(256)
void logsoftmax40_kernel(float* __restrict__ out, int n) {
  const int row = (int)((blockIdx.x * (size_t)blockDim.x + threadIdx.x) >> 5);
  const int lane = threadIdx.x & 31;
  if (row >= n) return;
  float* p = out + (size_t)row * ODIM;
  float a = -INFINITY, b = -INFINITY;
  if (lane < 20) { a = p[2 * lane]; b = p[2 * lane + 1]; }
  float m = fmaxf(a, b);
#pragma unroll
  for (int off = 16; off > 0; off >>= 1) m = fmaxf(m, __shfl_xor(m, off, 32));
  float s = (lane < 20) ? (expf(a - m) + expf(b - m)) : 0.f;
#pragma unroll
  for (int off = 16; off > 0; off >>= 1) s += __shfl_xor(s, off, 32);
  const float lse = m + logf(s);
  if (lane < 20) {
    p[2 * lane] = a - lse;
    p[2 * lane + 1] = b - lse;
  }
}

// ------------------------------------------------------------------
// launch
// ------------------------------------------------------------------
extern "C" void kernel_launch(void* const* d_in, const int* in_sizes, int n_in,
                              void* d_out, int out_size, void* d_ws, size_t ws_size,
                              hipStream_t stream) {
  (void)in_sizes; (void)n_in; (void)out_size; (void)ws_size;
  const float* x   = (const float*)d_in[0];
  const int*   ei  = (const int*)d_in[1];
  const float* W0  = (const float*)d_in[2];
  const float* b0  = (const float*)d_in[3];
  const float* W1  = (const float*)d_in[4];
  const float* b1  = (const float*)d_in[5];
  const float* W2  = (const float*)d_in[6];
  const float* b2  = (const float*)d_in[7];
  const float* g0  = (const float*)d_in[8];
  const float* be0 = (const float*)d_in[9];
  const float* g1  = (const float*)d_in[10];
  const float* be1 = (const float*)d_in[11];
  float* out = (float*)d_out;

  const int* rowIdx = ei;             // edge_index[0] : source j (gather)
  const int* colIdx = ei + N_EDGESC;  // edge_index[1] : target i (scatter)

  // ---- workspace carve ----
  char* ws = (char*)d_ws;
  size_t off = 0;
  auto carve = [&](size_t bytes) -> void* {
    void* p = ws + off;
    off += (bytes + 255) & ~(size_t)255;
    return p;
  };
  float*  dinv   = (float*)carve((size_t)N_NODES * 4);
  double* bnsum  = (double*)carve(FDIM * 8);
  double* bnsq   = (double*)carve(FDIM * 8);
  float*  bscale = (float*)carve(FDIM * 4);
  float*  bshift = (float*)carve(FDIM * 4);
  float*  bufA   = (float*)carve((size_t)N_NODES * FDIM * 4);
  float*  bufB   = (float*)carve((size_t)N_NODES * FDIM * 4);

  const int mtiles = (N_NODES + 15) / 16;
  const size_t totalH = (size_t)N_NODES * FDIM;
  const int gridElemH = (int)((totalH + 255) / 256);
  const int gridElemO = (int)(((size_t)N_NODES * ODIM + 255) / 256);
  const int gridEdgeW = (N_EDGESC + 7) / 8;    // 8 waves / 256-thread block, 1 edge per wave
  const int rowsPerBlk = 256;
  const int gridStats = (N_NODES + rowsPerBlk - 1) / rowsPerBlk;

  // ---- degree -> dinv (shared by all three convs) ----
  init_deg_kernel<<<(N_NODES + 255) / 256, 256, 0, stream>>>(dinv, N_NODES);
  degree_kernel<<<(N_EDGESC + 255) / 256, 256, 0, stream>>>(colIdx, dinv, N_EDGESC);
  dinv_kernel<<<(N_NODES + 255) / 256, 256, 0, stream>>>(dinv, N_NODES);

  // ---- conv0: bufA = x @ W0 ; bufB = aggregate(bufA) + b0 ; BN0+ReLU in place ----
  gemm_wmma_f32<FDIM, 8><<<mtiles, 256, 0, stream>>>(x, W0, bufA, N_NODES, FDIM);
  hipMemsetAsync(bufB, 0, totalH * 4, stream);
  scatter128_kernel<<<gridEdgeW, 256, 0, stream>>>(rowIdx, colIdx, dinv, bufA, bufB, N_EDGESC);
  selfloop_bias_kernel<FDIM><<<gridElemH, 256, 0, stream>>>(bufA, dinv, b0, bufB, N_NODES);
  hipMemsetAsync(bnsum, 0, FDIM * 16, stream);
  bn_stats_kernel<<<gridStats, FDIM, 0, stream>>>(bufB, bnsum, bnsq, N_NODES, rowsPerBlk);
  bn_finalize_kernel<<<1, FDIM, 0, stream>>>(bnsum, bnsq, g0, be0, bscale, bshift, N_NODES);
  bn_relu_kernel<<<gridElemH, 256, 0, stream>>>(bufB, bscale, bshift, totalH);

  // ---- conv1: bufA = bufB @ W1 ; bufB = aggregate(bufA) + b1 ; BN1+ReLU ----
  gemm_wmma_f32<FDIM, 8><<<mtiles, 256, 0, stream>>>(bufB, W1, bufA, N_NODES, FDIM);
  hipMemsetAsync(bufB, 0, totalH * 4, stream);
  scatter128_kernel<<<gridEdgeW, 256, 0, stream>>>(rowIdx, colIdx, dinv, bufA, bufB, N_EDGESC);
  selfloop_bias_kernel<FDIM><<<gridElemH, 256, 0, stream>>>(bufA, dinv, b1, bufB, N_NODES);
  hipMemsetAsync(bnsum, 0, FDIM * 16, stream);
  bn_stats_kernel<<<gridStats, FDIM, 0, stream>>>(bufB, bnsum, bnsq, N_NODES, rowsPerBlk);
  bn_finalize_kernel<<<1, FDIM, 0, stream>>>(bnsum, bnsq, g1, be1, bscale, bshift, N_NODES);
  bn_relu_kernel<<<gridElemH, 256, 0, stream>>>(bufB, bscale, bshift, totalH);

  // ---- conv2: bufA = bufB @ W2 (N x 40) ; aggregate into d_out ; log_softmax in place ----
  gemm_wmma_f32<ODIM, 3><<<mtiles, 96, 0, stream>>>(bufB, W2, bufA, N_NODES, FDIM);
  hipMemsetAsync(out, 0, (size_t)N_NODES * ODIM * 4, stream);
  scatter40_kernel<<<gridEdgeW, 256, 0, stream>>>(rowIdx, colIdx, dinv, bufA, out, N_EDGESC);
  selfloop_bias_kernel<ODIM><<<gridElemO, 256, 0, stream>>>(bufA, dinv, b2, out, N_NODES);
  logsoftmax40_kernel<<<(N_NODES + 7) / 8, 256, 0, stream>>>(out, N_NODES);
}